// LocalGlobalAttention_42331197669517
// MI455X (gfx1250) — compile-verified
//
#include <hip/hip_runtime.h>

// Shapes (fixed by the reference)
#define NTOK 2048
#define DMODEL 1024
#define NHEADS 16
#define KVHEADS 4
#define HDIM 64
#define KVDIM 256   // KVHEADS * HDIM
#define WIN 64
#define NGLOBAL 32  // NTOK / WIN
#define SCALE 0.125f

typedef float v2f __attribute__((ext_vector_type(2)));
typedef float v8f __attribute__((ext_vector_type(8)));

// ---------------------------------------------------------------------------
// Fragment loaders / MMA step for the 64x32-per-wave fp32 WMMA GEMM.
// WMMA f32 16x16x4 operand layout (wave32), h = lane>>4, l = lane&15:
//   A (16x4):  lane (h,l) holds A[l][k0+2h], A[l][k0+2h+1] in v0,v1
//   B (4x16):  lane (h,l) holds B[k0+2h][n0+l], B[k0+2h+1][n0+l] in v0,v1
//   C (16x16): VGPR r, lane (h,l) -> C[r + 8h][l]
// ---------------------------------------------------------------------------
__device__ __forceinline__ void load_frag(const float* __restrict__ ap,
                                          const float* __restrict__ bp,
                                          size_t aRowTile, int N,
                                          v2f a[4], v2f b[2]) {
#pragma unroll
  for (int mi = 0; mi < 4; ++mi)
    a[mi] = *(const v2f*)(ap + (size_t)mi * aRowTile);  // 8B aligned
#pragma unroll
  for (int ni = 0; ni < 2; ++ni) {
    v2f bv;
    bv.x = bp[ni * 16];
    bv.y = bp[ni * 16 + N];
    b[ni] = bv;
  }
}

__device__ __forceinline__ void mma_step(v8f acc[4][2], const v2f a[4],
                                         const v2f b[2]) {
#pragma unroll
  for (int mi = 0; mi < 4; ++mi)
#pragma unroll
    for (int ni = 0; ni < 2; ++ni)
      acc[mi][ni] = __builtin_amdgcn_wmma_f32_16x16x4_f32(
          false, a[mi], false, b[ni], (short)0, acc[mi][ni], false, false);
}

// ---------------------------------------------------------------------------
// Row-major fp32 GEMM: C[M,N] = A[M,K] * B[K,N] using V_WMMA_F32_16X16X4_F32.
// Block = 128 threads = 4 waves (2x2) covering 128x64 of C; each wave owns a
// 64x32 region (4x2 16x16 accumulators, 8 WMMAs per K-step of 4). The K loop
// is software-pipelined with double-buffered fragments: loads for chunk n+1
// are in flight while chunk n's 8 WMMAs issue, with a peeled epilogue so no
// prefetch reads past K. Requires M%128==0, N%64==0, K%8==0, K>=16.
// ---------------------------------------------------------------------------
__global__ __launch_bounds__(128) void gemm_f32_wmma(
    const float* __restrict__ A, const float* __restrict__ B,
    float* __restrict__ C, int M, int N, int K) {
  const int wave = threadIdx.x >> 5;
  const int lane = threadIdx.x & 31;
  const int half = lane >> 4;   // 0: K pair {k0,k0+1}, 1: {k0+2,k0+3}
  const int l    = lane & 15;

  const int m0 = blockIdx.y * 128 + (wave >> 1) * 64;
  const int n0 = blockIdx.x * 64 + (wave & 1) * 32;

  v8f acc[4][2] = {};

  const float* ap = A + (size_t)(m0 + l) * K + 2 * half;
  const float* bp = B + (size_t)(2 * half) * N + (n0 + l);
  const size_t aRowTile = (size_t)16 * K;  // 16 M-rows
  const size_t bStep = (size_t)4 * N;      // 4 K-rows

  v2f a0[4], b0[2], a1[4], b1[2];

  // Prologue: chunk at k=0 in flight before the loop body starts.
  load_frag(ap, bp, aRowTile, N, a0, b0);

  for (int k0 = 0; k0 < K - 8; k0 += 8) {
    // Issue chunk k0+4, consume chunk k0.
    load_frag(ap + 4, bp + bStep, aRowTile, N, a1, b1);
    mma_step(acc, a0, b0);
    ap += 8;
    bp += 2 * bStep;
    // Issue chunk k0+8, consume chunk k0+4.
    load_frag(ap, bp, aRowTile, N, a0, b0);
    mma_step(acc, a1, b1);
  }
  // Epilogue: chunks K-8 (already in a0/b0) and K-4 (no further prefetch).
  load_frag(ap + 4, bp + bStep, aRowTile, N, a1, b1);
  mma_step(acc, a0, b0);
  mma_step(acc, a1, b1);

#pragma unroll
  for (int mi = 0; mi < 4; ++mi)
#pragma unroll
    for (int ni = 0; ni < 2; ++ni)
#pragma unroll
      for (int r = 0; r < 8; ++r)
        C[(size_t)(m0 + 16 * mi + r + 8 * half) * N + (n0 + 16 * ni + l)] =
            acc[mi][ni][r];
}

// ---------------------------------------------------------------------------
// Banded-causal ALiBi attention (local window) + strided global attention.
// One wave32 per (query i, head h). Each lane owns 2 of the 64 head dims.
// Online softmax; butterfly shfl_xor reduction for q.k dot products.
// Q layout: [tok][h*64+d]; K/V layout: [tok][kvh*64+d]; O: [tok][h*64+d].
// ---------------------------------------------------------------------------
__global__ __launch_bounds__(256) void attn_local_global(
    const float* __restrict__ Q, const float* __restrict__ K,
    const float* __restrict__ V, float* __restrict__ O) {
  const int wid  = blockIdx.x * (blockDim.x >> 5) + (threadIdx.x >> 5);
  const int lane = threadIdx.x & 31;
  const int i = wid >> 4;   // query token
  const int h = wid & 15;   // head
  if (i >= NTOK) return;
  const int kvh = h >> 2;   // GQA: 4 query heads per KV head

  const float* q = Q + (size_t)i * DMODEL + h * HDIM;
  const float q0 = q[lane * 2];
  const float q1 = q[lane * 2 + 1];
  const float slope = exp2f(-0.5f * (float)(h + 1));  // alibi slopes, 16 heads

  // ----- local pass: j in [max(0, i-64), i], score = q.k*SCALE + slope*(j-i)
  float m = -1e30f, s = 0.f, a0 = 0.f, a1 = 0.f;
  int j0 = i - WIN;
  if (j0 < 0) j0 = 0;
  for (int j = j0; j <= i; ++j) {
    const float* kp = K + (size_t)j * KVDIM + kvh * HDIM;
    float p = q0 * kp[lane * 2] + q1 * kp[lane * 2 + 1];
#pragma unroll
    for (int off = 16; off > 0; off >>= 1) p += __shfl_xor(p, off, 32);
    const float score = p * SCALE + slope * (float)(j - i);
    const float mnew = fmaxf(m, score);
    const float resc = __expf(m - mnew);   // 0 on first iteration
    const float w = __expf(score - mnew);
    const float* vp = V + (size_t)j * KVDIM + kvh * HDIM;
    s  = s * resc + w;
    a0 = a0 * resc + w * vp[lane * 2];
    a1 = a1 * resc + w * vp[lane * 2 + 1];
    m = mnew;
  }
  const float invs = 1.0f / s;
  float o0 = a0 * invs;
  float o1 = a1 * invs;

  // ----- global pass: 32 keys at j = g*64, plain softmax (no mask, no alibi)
  float mg = -1e30f, sg = 0.f, g0 = 0.f, g1 = 0.f;
  for (int g = 0; g < NGLOBAL; ++g) {
    const int j = g * WIN;
    const float* kp = K + (size_t)j * KVDIM + kvh * HDIM;
    float p = q0 * kp[lane * 2] + q1 * kp[lane * 2 + 1];
#pragma unroll
    for (int off = 16; off > 0; off >>= 1) p += __shfl_xor(p, off, 32);
    const float score = p * SCALE;
    const float mnew = fmaxf(mg, score);
    const float resc = __expf(mg - mnew);
    const float w = __expf(score - mnew);
    const float* vp = V + (size_t)j * KVDIM + kvh * HDIM;
    sg = sg * resc + w;
    g0 = g0 * resc + w * vp[lane * 2];
    g1 = g1 * resc + w * vp[lane * 2 + 1];
    mg = mnew;
  }
  const float invg = 1.0f / sg;
  o0 += g0 * invg;
  o1 += g1 * invg;

  float* op = O + (size_t)i * DMODEL + h * HDIM;
  op[lane * 2]     = o0;
  op[lane * 2 + 1] = o1;
}

// ---------------------------------------------------------------------------
// Host-side launcher. Inputs: x, Wq, Wk, Wv, Wo (all fp32). Output: fp32.
// Workspace (fp32): Q[2048*1024] | K[2048*256] | V[2048*256] | ATT[2048*1024]
// = 20 MB total.
// ---------------------------------------------------------------------------
extern "C" void kernel_launch(void* const* d_in, const int* in_sizes, int n_in,
                              void* d_out, int out_size, void* d_ws, size_t ws_size,
                              hipStream_t stream) {
  const float* x  = (const float*)d_in[0];
  const float* Wq = (const float*)d_in[1];
  const float* Wk = (const float*)d_in[2];
  const float* Wv = (const float*)d_in[3];
  const float* Wo = (const float*)d_in[4];
  float* out = (float*)d_out;

  float* Q   = (float*)d_ws;
  float* Kb  = Q  + (size_t)NTOK * DMODEL;
  float* Vb  = Kb + (size_t)NTOK * KVDIM;
  float* ATT = Vb + (size_t)NTOK * KVDIM;

  const dim3 gblk(128, 1, 1);

  // Projections (grid = (N/64, M/128))
  gemm_f32_wmma<<<dim3(DMODEL / 64, NTOK / 128), gblk, 0, stream>>>(
      x, Wq, Q, NTOK, DMODEL, DMODEL);
  gemm_f32_wmma<<<dim3(KVDIM / 64, NTOK / 128), gblk, 0, stream>>>(
      x, Wk, Kb, NTOK, KVDIM, DMODEL);
  gemm_f32_wmma<<<dim3(KVDIM / 64, NTOK / 128), gblk, 0, stream>>>(
      x, Wv, Vb, NTOK, KVDIM, DMODEL);

  // Attention: one wave per (token, head) -> 2048*16 waves, 8 waves/block
  attn_local_global<<<dim3((NTOK * NHEADS) / 8), dim3(256), 0, stream>>>(
      Q, Kb, Vb, ATT);

  // Output projection
  gemm_f32_wmma<<<dim3(DMODEL / 64, NTOK / 128), gblk, 0, stream>>>(
      ATT, Wo, out, NTOK, DMODEL, DMODEL);
}